// LocalSpatialEncoding_79774722556531
// MI455X (gfx1250) — compile-verified
//
#include <hip/hip_runtime.h>
#include <hip/hip_bf16.h>

typedef __attribute__((ext_vector_type(2))) float v2f;
typedef __attribute__((ext_vector_type(8))) float v8f;

// Problem constants (from reference setup_inputs)
constexpr int B_ = 4;
constexpr int N_ = 40960;
constexpr int K_ = 16;
constexpr int D_ = 8;     // conv output channels
constexpr float EPS_ = 1e-6f;

// d_ws float layout:
//   [0..7]    sum_d(h)
//   [8..15]   sum_d(h^2)
//   [16..207] WMMA B-operand (3 K-chunks x 2 VGPRs x 32 lanes) of BN-folded W
//   [208..223] folded bias per column lane (d<8 valid)

// ---------------------------------------------------------------- pass 0
__global__ void locse_zero(float* ws) {
    int t = threadIdx.x;
    if (t < 16) ws[t] = 0.0f;
}

// ---------------------------------------------------------------- pass 1: batch stats
__global__ __launch_bounds__(256) void locse_stats(const float* __restrict__ coords,
                                                   const int* __restrict__ idx,
                                                   const float* __restrict__ dist,
                                                   const float* __restrict__ W,
                                                   const float* __restrict__ bvec,
                                                   float* __restrict__ ws) {
    __shared__ float sW[80];
    __shared__ float sb[8];
    __shared__ float sacc[16];
    int tid = threadIdx.x;
    if (tid < 80) sW[tid] = W[tid];
    if (tid < 8)  sb[tid] = bvec[tid];
    if (tid < 16) sacc[tid] = 0.0f;
    __syncthreads();

    float sum[8] = {0,0,0,0,0,0,0,0};
    float sq [8] = {0,0,0,0,0,0,0,0};

    const int total = B_ * N_ * K_;                 // 2,621,440
    for (int p = blockIdx.x * blockDim.x + tid; p < total;
         p += gridDim.x * blockDim.x) {
        unsigned b = (unsigned)p / (unsigned)(N_ * K_);
        unsigned r = (unsigned)p - b * (unsigned)(N_ * K_);
        unsigned n = r >> 4;                        // K_ == 16
        int j = idx[p];
        float dd = dist[p];
        const float* cp = coords + 3u * (b * N_ + n);
        float cx = cp[0], cy = cp[1], cz = cp[2];
        const float* qp = coords + 3u * (b * N_ + (unsigned)j);
        float nx = qp[0], ny = qp[1], nz = qp[2];
        float c10[10] = {cx, cy, cz, nx, ny, nz, cx - nx, cy - ny, cz - nz, dd};
#pragma unroll
        for (int d = 0; d < 8; ++d) {
            float h = sb[d];
#pragma unroll
            for (int c = 0; c < 10; ++c) h = fmaf(c10[c], sW[d * 10 + c], h);
            sum[d] += h;
            sq[d]  += h * h;
        }
    }
#pragma unroll
    for (int d = 0; d < 8; ++d) {
        atomicAdd(&sacc[d], sum[d]);
        atomicAdd(&sacc[8 + d], sq[d]);
    }
    __syncthreads();
    if (tid < 16) atomicAdd(&ws[tid], sacc[tid]);
}

// ---------------------------------------------------------------- pass 2: fold BN into W,b
// B-operand layout (mirrors documented 32-bit A 16x4 layout):
// chunk t, VGPR j, lane L: K-row = 4t + j + 2*(L>=16), column d = L&15.
__global__ void locse_prep(const float* __restrict__ W,
                           const float* __restrict__ bvec,
                           const float* __restrict__ gamma,
                           const float* __restrict__ beta,
                           float* __restrict__ ws) {
    int lane = threadIdx.x;                         // 0..31
    int dl = lane & 15;
    int hi = lane >> 4;
    const float M = (float)(B_ * N_ * K_);
    float a = 0.0f, bias = 0.0f;
    if (dl < 8) {
        float s  = ws[dl];
        float sq = ws[8 + dl];
        float mean = s / M;
        float var  = sq / M - mean * mean;
        float rs   = rsqrtf(var + EPS_);
        a    = gamma[dl] * rs;
        bias = a * (bvec[dl] - mean) + beta[dl];
    }
#pragma unroll
    for (int t = 0; t < 3; ++t)
#pragma unroll
        for (int j = 0; j < 2; ++j) {
            int c = 4 * t + j + 2 * hi;
            float v = (dl < 8 && c < 10) ? a * W[dl * 10 + c] : 0.0f;
            ws[16 + (t * 2 + j) * 32 + lane] = v;
        }
    if (lane < 16) ws[208 + lane] = bias;
}

// ---------------------------------------------------------------- pass 3: main output (WMMA)
// One wave per (b,n): A rows = 16 neighbors, B = folded 10x8 weights, 3x K=4 chunks.
// Fully branchless body: EXEC stays all-1s (WMMA requirement), stores are 2x b128/lane.
__global__ __launch_bounds__(256) void locse_main(const float* __restrict__ coords,
                                                  const float* __restrict__ features,
                                                  const int* __restrict__ idx,
                                                  const float* __restrict__ dist,
                                                  const float* __restrict__ ws,
                                                  float* __restrict__ out) {
    const float* wsB = ws + 16;
    const float* wsbias = ws + 208;
    int lane = threadIdx.x & 31;
    int wave = threadIdx.x >> 5;
    unsigned dl = (unsigned)(lane & 15);
    unsigned hi = (unsigned)(lane >> 4);
    bool isH = dl < 8;

    v2f B0, B1, B2;
    B0.x = wsB[lane];       B0.y = wsB[32 + lane];
    B1.x = wsB[64 + lane];  B1.y = wsB[96 + lane];
    B2.x = wsB[128 + lane]; B2.y = wsB[160 + lane];
    float bias = wsbias[dl];

    int w = blockIdx.x * 8 + wave;
    int nwaves = gridDim.x * 8;
    for (int t = w; t < B_ * N_; t += nwaves) {
        unsigned b = (unsigned)t / (unsigned)N_;
        unsigned n = (unsigned)t - b * (unsigned)N_;
        unsigned p = (b * N_ + n) * 16u + dl;       // neighbor slot for this lane
        int j = idx[p];
        float dd = dist[p];
        const float* cp = coords + 3u * (b * N_ + n);
        float cx = cp[0], cy = cp[1], cz = cp[2];
        const float* qp = coords + 3u * (b * N_ + (unsigned)j);
        float nx = qp[0], ny = qp[1], nz = qp[2];

        // Broadcast-feature value for the padding-column lanes (clamped channel;
        // lanes 0..7 read the same 32B as 8..15 -> coalesced, L2-resident).
        float f = features[(b * (unsigned)D_ + (dl & 7u)) * (unsigned)N_ + n];

        v2f A0, A1, A2;
        A0.x = hi ? cz : cx;
        A0.y = hi ? nx : cy;
        A1.x = hi ? (cx - nx) : ny;
        A1.y = hi ? (cy - ny) : nz;
        A2.x = hi ? 0.0f : (cz - nz);
        A2.y = hi ? 0.0f : dd;

        v8f acc = {bias, bias, bias, bias, bias, bias, bias, bias};
        acc = __builtin_amdgcn_wmma_f32_16x16x4_f32(false, A0, false, B0,
                                                    (short)0, acc, false, false);
        acc = __builtin_amdgcn_wmma_f32_16x16x4_f32(false, A1, false, B1,
                                                    (short)0, acc, false, false);
        acc = __builtin_amdgcn_wmma_f32_16x16x4_f32(false, A2, false, B2,
                                                    (short)0, acc, false, false);

        // Output [B, 16, N, 16]: lanes dl<8 write ReLU(BN(conv)) for channel dl,
        // k = hi*8..hi*8+7 (C/D VGPR i = row k = i + 8*hi). Lanes dl>=8 write the
        // broadcast features half. Branchless select keeps EXEC uniform.
        float4 lo4, hi4;
        lo4.x = isH ? fmaxf(acc[0], 0.0f) : f;
        lo4.y = isH ? fmaxf(acc[1], 0.0f) : f;
        lo4.z = isH ? fmaxf(acc[2], 0.0f) : f;
        lo4.w = isH ? fmaxf(acc[3], 0.0f) : f;
        hi4.x = isH ? fmaxf(acc[4], 0.0f) : f;
        hi4.y = isH ? fmaxf(acc[5], 0.0f) : f;
        hi4.z = isH ? fmaxf(acc[6], 0.0f) : f;
        hi4.w = isH ? fmaxf(acc[7], 0.0f) : f;

        unsigned obase = ((b * 16u + dl) * (unsigned)N_ + n) * 16u + hi * 8u;
        float4* op = (float4*)(out + obase);
        op[0] = lo4;
        op[1] = hi4;
    }
}

extern "C" void kernel_launch(void* const* d_in, const int* in_sizes, int n_in,
                              void* d_out, int out_size, void* d_ws, size_t ws_size,
                              hipStream_t stream) {
    const float* coords   = (const float*)d_in[0];
    const float* features = (const float*)d_in[1];
    const int*   idx      = (const int*)d_in[2];
    const float* dist     = (const float*)d_in[3];
    const float* W        = (const float*)d_in[4];
    const float* bvec     = (const float*)d_in[5];
    const float* gamma    = (const float*)d_in[6];
    const float* beta     = (const float*)d_in[7];
    float* out = (float*)d_out;
    float* ws  = (float*)d_ws;

    locse_zero<<<1, 32, 0, stream>>>(ws);
    locse_stats<<<1024, 256, 0, stream>>>(coords, idx, dist, W, bvec, ws);
    locse_prep<<<1, 32, 0, stream>>>(W, bvec, gamma, beta, ws);
    locse_main<<<2048, 256, 0, stream>>>(coords, features, idx, dist, ws, out);
}